// ComplexFruitFly_54795192762760
// MI455X (gfx1250) — compile-verified
//
#include <hip/hip_runtime.h>
#include <stdint.h>

#define K_DIM 1024
#define V_DIM 50000
#define B_DIM 2048
#define WD_DIM 20
#define THRF 1e-10f

#define KT 256   // k-tile per block (gridDim.y slow axis -> L2-resident k-window of 100MB)
#define BT 16    // b-tile per block (gridDim.x fast axis)

typedef __attribute__((ext_vector_type(2))) float v2f;
typedef __attribute__((ext_vector_type(8))) float v8f;

__device__ __forceinline__ float wave_sum_f32(float v) {
#pragma unroll
  for (int off = 16; off > 0; off >>= 1) v += __shfl_xor(v, off, 32);
  return v;
}

__device__ __forceinline__ unsigned long long wave_max_u64(unsigned long long v) {
#pragma unroll
  for (int off = 16; off > 0; off >>= 1) {
    unsigned int lo = (unsigned int)(v & 0xffffffffull);
    unsigned int hi = (unsigned int)(v >> 32);
    unsigned int olo = __shfl_xor(lo, off, 32);
    unsigned int ohi = __shfl_xor(hi, off, 32);
    unsigned long long o = (((unsigned long long)ohi) << 32) | (unsigned long long)olo;
    v = (o > v) ? o : v;
  }
  return v;
}

// ---------------------------------------------------------------------------
// Kernel A: per-k row norms via WMMA Gram-diagonal, plus workspace zeroing.
// One wave per k-row. Row = 2*V = 100000 contiguous floats. Tiles of 64
// floats feed a 16x4 f32 A-operand; since B-layout(A^T) == A-layout(A)
// register-for-register for a symmetric product, we pass `a` twice and the
// accumulator's diagonal collects per-segment sums of squares.
// ---------------------------------------------------------------------------
__global__ void norms_kernel(const float* __restrict__ W,
                             unsigned long long* __restrict__ ws_mu,
                             float* __restrict__ ws_den) {
  int gid = blockIdx.x * blockDim.x + threadIdx.x;
  if (gid < B_DIM) ws_mu[gid] = 0ull;  // scores are >0, so 0 is a safe identity

  const int k    = gid >> 5;           // 128 blocks * 8 waves = 1024 rows
  const int lane = threadIdx.x & 31;
  const int ROWF = 2 * V_DIM;          // 100000 floats per row
  const float* row = W + (size_t)k * (size_t)ROWF;

  // A-operand lane offsets within a 64-float tile:
  // VGPR0: lanes0-15 -> col0 (floats 0..15), lanes16-31 -> col2 (floats 32..47)
  // VGPR1: lanes0-15 -> col1 (floats16..31), lanes16-31 -> col3 (floats 48..63)
  const int o0 = (lane & 16) ? ((lane & 15) + 32) : lane;
  const int o1 = o0 + 16;

  v8f c = {};
  const int NT = ROWF / 64;  // 1562 full tiles
  for (int t = 0; t < NT; ++t) {
    const float* p = row + t * 64;
    v2f a;
    a.x = p[o0];
    a.y = p[o1];
    c = __builtin_amdgcn_wmma_f32_16x16x4_f32(false, a, false, a, (short)0, c,
                                              false, false);
  }
  {  // tail: last 32 floats, guarded
    int base = NT * 64;
    v2f a;
    a.x = (base + o0 < ROWF) ? row[base + o0] : 0.f;
    a.y = (base + o1 < ROWF) ? row[base + o1] : 0.f;
    c = __builtin_amdgcn_wmma_f32_16x16x4_f32(false, a, false, a, (short)0, c,
                                              false, false);
  }

  // Extract trace(D): D[m][m] lives at (vgpr j, lane j) for m=j and
  // (vgpr j, lane 24+j) for m=8+j.
  float contrib = 0.f;
#pragma unroll
  for (int j = 0; j < 8; ++j) {
    if (lane == j)      contrib = c[j];
    if (lane == 24 + j) contrib = c[j];
  }
  float tot = wave_sum_f32(contrib);
  if (lane == 0) ws_den[k] = __builtin_sqrtf(tot);
}

// ---------------------------------------------------------------------------
// Kernel B: scores + argmax. One thread per k, BT batch rows per block.
// gridDim = (B/BT fast, K/KT slow) so resident blocks share one k-window
// (KT rows = 100MB) and gather reuse is served from the 192MB L2.
// ---------------------------------------------------------------------------
__global__ void scores_kernel(const float* __restrict__ W,
                              const float* __restrict__ pos,
                              const int* __restrict__ ids,
                              unsigned long long* __restrict__ ws_mu) {
  __shared__ int   s_id[BT][WD_DIM];
  __shared__ float s_a[BT][WD_DIM];
  const float PI = 3.14159265358979323846f;
  const float TWO_PI = 6.28318530717958647692f;

  const int t = threadIdx.x;
  for (int i = t; i < BT * WD_DIM; i += KT) {
    int bb = i / WD_DIM, w = i % WD_DIM;
    int b = blockIdx.x * BT + bb;
    s_id[bb][w] = ids[b * WD_DIM + w];
    s_a[bb][w]  = PI * pos[b * WD_DIM + w];
  }
  __syncthreads();

  const int k = blockIdx.y * KT + t;
  const float2* Wrow = reinterpret_cast<const float2*>(W) + (size_t)k * V_DIM;

  for (int bb = 0; bb < BT; ++bb) {
    float s = 0.f;
#pragma unroll
    for (int w = 0; w < WD_DIM; ++w) {
      float2 g = Wrow[s_id[bb][w]];                       // 8B gather, L2-resident row
      float mag = __builtin_sqrtf(g.x * g.x + g.y * g.y); // |g|
      float ang = atan2f(g.y, g.x);                       // angle(g)
      float d = fabsf(ang - s_a[bb][w]);                  // |angle(g) - pi*pos|
      if (d > PI) d = TWO_PI - d;                         // wrap to (-pi, pi]
      s += mag + d;
    }
    // pack (score, k); max over wave then one atomic per wave per b.
    unsigned long long key =
        (((unsigned long long)__float_as_uint(s)) << 32) | (unsigned int)k;
    key = wave_max_u64(key);
    if ((t & 31) == 0) {
      int b = blockIdx.x * BT + bb;
      atomicMax(&ws_mu[b], key);  // max is exactly associative -> deterministic
    }
  }
}

// ---------------------------------------------------------------------------
// Kernel C: per-b energy partials + mu output. One wave per batch row.
// ---------------------------------------------------------------------------
__global__ void energy_kernel(const float* __restrict__ W,
                              const float* __restrict__ Ps,
                              const float* __restrict__ pos,
                              const int* __restrict__ ids,
                              const unsigned long long* __restrict__ ws_mu,
                              const float* __restrict__ ws_den,
                              float* __restrict__ ws_Eb,
                              float* __restrict__ d_out) {
  const int b = blockIdx.x;
  const int t = threadIdx.x;  // 0..31
  const unsigned int mu = (unsigned int)(ws_mu[b] & 0xffffffffull);

  float num = 0.f, aphi = 0.f;
  if (t < WD_DIM) {
    int   id = ids[b * WD_DIM + t];
    float P  = Ps[b * WD_DIM + t];
    float th = 3.14159265358979323846f * pos[b * WD_DIM + t];
    float2 g = reinterpret_cast<const float2*>(W)[(size_t)mu * V_DIM + id];
    float cs = cosf(th), sn = sinf(th);
    // alpha = Wc * Ps * conj(e^{i*th})
    float ar = P * (g.x * cs + g.y * sn);
    float ai = P * (g.y * cs - g.x * sn);
    if (ar < THRF && ar > -THRF) ar = THRF;
    if (ai < THRF && ai > -THRF) ai = THRF;
    num  = __builtin_sqrtf(ar * ar + ai * ai);
    aphi = fabsf(atan2f(ai, ar)) * P;
  }
  num  = wave_sum_f32(num);
  aphi = wave_sum_f32(aphi);
  if (t == 0) {
    d_out[b] = (float)mu;                       // mu output (exact in f32)
    ws_Eb[b] = -(num / ws_den[mu]) - aphi;      // per-b energy partial
  }
}

// ---------------------------------------------------------------------------
// Kernel D: fixed-order reduction of 2048 energy partials -> E (deterministic).
// ---------------------------------------------------------------------------
__global__ void finalize_kernel(const float* __restrict__ ws_Eb,
                                float* __restrict__ d_out) {
  __shared__ float s[256];
  const int t = threadIdx.x;
  float v = 0.f;
#pragma unroll
  for (int i = 0; i < B_DIM / 256; ++i) v += ws_Eb[t + i * 256];
  s[t] = v;
  __syncthreads();
  for (int off = 128; off > 0; off >>= 1) {
    if (t < off) s[t] += s[t + off];
    __syncthreads();
  }
  if (t == 0) d_out[B_DIM] = s[0];
}

extern "C" void kernel_launch(void* const* d_in, const int* in_sizes, int n_in,
                              void* d_out, int out_size, void* d_ws, size_t ws_size,
                              hipStream_t stream) {
  const float* W   = (const float*)d_in[0];  // (K, V, 2)
  const float* Ps  = (const float*)d_in[1];  // (B, WD)
  const float* pos = (const float*)d_in[2];  // (B, WD)
  const int*   ids = (const int*)d_in[3];    // (B, WD)
  // d_in[4] = top_k (always 1 per setup_inputs)

  unsigned long long* ws_mu = (unsigned long long*)d_ws;                 // 16 KB
  float* ws_den = (float*)((char*)d_ws + (size_t)B_DIM * 8);             //  4 KB
  float* ws_Eb  = (float*)((char*)d_ws + (size_t)B_DIM * 8 + K_DIM * 4); //  8 KB
  float* out = (float*)d_out;  // [0..B) = mu, [B] = E

  norms_kernel<<<dim3(K_DIM / 8), dim3(256), 0, stream>>>(W, ws_mu, ws_den);
  scores_kernel<<<dim3(B_DIM / BT, K_DIM / KT), dim3(KT), 0, stream>>>(W, pos, ids, ws_mu);
  energy_kernel<<<dim3(B_DIM), dim3(32), 0, stream>>>(W, Ps, pos, ids, ws_mu,
                                                      ws_den, ws_Eb, out);
  finalize_kernel<<<dim3(1), dim3(256), 0, stream>>>(ws_Eb, out);
}